// MeanGraphLayer_63909113365223
// MI455X (gfx1250) — compile-verified
//
#include <hip/hip_runtime.h>
#include <math.h>

#define N_NODES 50000
#define N_EDGES 800000
#define DIM     64
#define TWO_DIM 128
#define HIDDEN  256
#define TILES   (N_NODES / 16)   // 3125 exact 16-row tiles
#define HSTR    260              // padded hid-row stride (floats) to dodge LDS bank conflicts

typedef float v2f __attribute__((ext_vector_type(2)));
typedef float v8f __attribute__((ext_vector_type(8)));

// ---------------------------------------------------------------- zero scratch
__global__ void gnn_zero_kernel(float4* __restrict__ p, int n4) {
    int i = blockIdx.x * blockDim.x + threadIdx.x;
    if (i < n4) p[i] = make_float4(0.f, 0.f, 0.f, 0.f);
}

// ------------------------------------------------------- edge scatter (atomics)
// 16 threads per edge; thread t handles dims [4t, 4t+4).
__global__ void gnn_scatter_kernel(const float* __restrict__ h,
                                   const int* __restrict__ ei,
                                   float* __restrict__ x,
                                   float* __restrict__ deg) {
    int gid = blockIdx.x * blockDim.x + threadIdx.x;
    int e = gid >> 4;
    int t = gid & 15;
    if (e >= N_EDGES) return;
    int s = ei[e];
    int d = ei[N_EDGES + e];
    float4 v = *(const float4*)(h + (size_t)s * DIM + t * 4);
    float* dst = x + (size_t)d * TWO_DIM + DIM + t * 4;
    atomicAdd(dst + 0, v.x);
    atomicAdd(dst + 1, v.y);
    atomicAdd(dst + 2, v.z);
    atomicAdd(dst + 3, v.w);
    if (t == 0) atomicAdd(deg + d, 1.0f);
}

// ------------------------------------------ build x = [h | mean (or h if deg==0)]
__global__ void gnn_finalize_kernel(const float* __restrict__ h,
                                    float* __restrict__ x,
                                    const float* __restrict__ deg) {
    int gid = blockIdx.x * blockDim.x + threadIdx.x;  // N_NODES*DIM threads
    int n = gid >> 6;
    int d = gid & 63;
    if (n >= N_NODES) return;
    float hv = h[(size_t)n * DIM + d];
    float s  = x[(size_t)n * TWO_DIM + DIM + d];
    float dg = deg[n];
    x[(size_t)n * TWO_DIM + d]       = hv;
    x[(size_t)n * TWO_DIM + DIM + d] = (dg > 0.0f) ? (s / dg) : hv;
}

// ------------------------------------------------- fused MLP: WMMA f32 16x16x4
// Block = 256 threads = 8 waves; wave w owns node tile (blockIdx.x*8 + w).
// LDS: W1 [128x256] f32 staged once per block, + per-wave 16xHSTR hid buffer.
__global__ __launch_bounds__(256)
void gnn_mlp_kernel(const float* __restrict__ x,
                    const float* __restrict__ W1,
                    const float* __restrict__ b1,
                    const float* __restrict__ W2,
                    const float* __restrict__ b2,
                    float* __restrict__ out) {
    extern __shared__ float lds[];
    float* W1s  = lds;                       // 128*256 floats
    float* hidS = lds + TWO_DIM * HIDDEN;    // 8 * 16 * HSTR floats

    const int tid  = threadIdx.x;
    const int wave = tid >> 5;
    const int lane = tid & 31;
    const int lr   = lane & 15;   // M (A/C) or N (B/C) sub-index
    const int lh   = lane >> 4;   // lane-half selector

    // stage W1 into LDS (coalesced float4)
    {
        const float4* src = (const float4*)W1;
        float4*       dst = (float4*)W1s;
        for (int i = tid; i < (TWO_DIM * HIDDEN) / 4; i += 256) dst[i] = src[i];
    }
    __syncthreads();

    const int tile = blockIdx.x * 8 + wave;
    if (tile >= TILES) return;
    const int row0 = tile * 16;

    // ---------------- GEMM1: hid = x[16x128] @ W1[128x256] + b1 ----------------
    v8f acc[16];
#pragma unroll
    for (int c = 0; c < 16; ++c) {
        float bv = b1[c * 16 + lr];
        acc[c] = (v8f){bv, bv, bv, bv, bv, bv, bv, bv};
    }
    const float* xrow = x + (size_t)(row0 + lr) * TWO_DIM + 2 * lh;
    for (int ks = 0; ks < TWO_DIM / 4; ++ks) {
        v2f a = *(const v2f*)(xrow + ks * 4);                    // A frag (16x4)
        const int k = ks * 4 + 2 * lh;
        const float* w0 = W1s + k * HIDDEN + lr;
        const float* w1 = w0 + HIDDEN;
#pragma unroll
        for (int c = 0; c < 16; ++c) {
            v2f b;
            b.x = w0[c * 16];                                    // B frag (4x16)
            b.y = w1[c * 16];
            acc[c] = __builtin_amdgcn_wmma_f32_16x16x4_f32(
                false, a, false, b, (short)0, acc[c], false, false);
        }
    }

    // exact GELU, stage hid tile to LDS (D-layout -> row-major)
    float* hbuf = hidS + wave * (16 * HSTR);
#pragma unroll
    for (int c = 0; c < 16; ++c) {
#pragma unroll
        for (int v = 0; v < 8; ++v) {
            const int m = v + 8 * lh;
            float val = acc[c][v];
            val = 0.5f * val * (1.0f + erff(val * 0.7071067811865475f));
            hbuf[m * HSTR + c * 16 + lr] = val;
        }
    }

    // ---------------- GEMM2: out = gelu(hid)[16x256] @ W2[256x64] + b2 ---------
    v8f acc2[4];
#pragma unroll
    for (int c = 0; c < 4; ++c) {
        float bv = b2[c * 16 + lr];
        acc2[c] = (v8f){bv, bv, bv, bv, bv, bv, bv, bv};
    }
    for (int ks = 0; ks < HIDDEN / 4; ++ks) {
        const int k = ks * 4 + 2 * lh;
        v2f a = *(const v2f*)(hbuf + lr * HSTR + k);             // A frag from LDS
        const float* w2p = W2 + (size_t)k * DIM + lr;
#pragma unroll
        for (int c = 0; c < 4; ++c) {
            v2f b;
            b.x = w2p[c * 16];
            b.y = w2p[DIM + c * 16];
            acc2[c] = __builtin_amdgcn_wmma_f32_16x16x4_f32(
                false, a, false, b, (short)0, acc2[c], false, false);
        }
    }

    // write out tile (D-layout)
#pragma unroll
    for (int c = 0; c < 4; ++c) {
#pragma unroll
        for (int v = 0; v < 8; ++v) {
            const int m = v + 8 * lh;
            out[(size_t)(row0 + m) * DIM + c * 16 + lr] = acc2[c][v];
        }
    }
}

// ---------------------------------------------------------------------- launch
extern "C" void kernel_launch(void* const* d_in, const int* in_sizes, int n_in,
                              void* d_out, int out_size, void* d_ws, size_t ws_size,
                              hipStream_t stream) {
    const float* h  = (const float*)d_in[0];
    const int*   ei = (const int*)d_in[1];   // [2, N_EDGES]: src row then dst row
    const float* W1 = (const float*)d_in[2];
    const float* b1 = (const float*)d_in[3];
    const float* W2 = (const float*)d_in[4];
    const float* b2 = (const float*)d_in[5];
    float* out = (float*)d_out;

    float* x   = (float*)d_ws;                       // [N_NODES, 128]
    float* deg = x + (size_t)N_NODES * TWO_DIM;      // [N_NODES]

    // 1) zero x + deg  (N*128 + N floats = divisible by 4)
    int n4 = (N_NODES * TWO_DIM + N_NODES) / 4;
    gnn_zero_kernel<<<(n4 + 255) / 256, 256, 0, stream>>>((float4*)x, n4);

    // 2) edge scatter
    gnn_scatter_kernel<<<(N_EDGES * 16) / 256, 256, 0, stream>>>(h, ei, x, deg);

    // 3) mean / isolated-node fixup, build concat input
    gnn_finalize_kernel<<<(N_NODES * DIM) / 256, 256, 0, stream>>>(h, x, deg);

    // 4) fused MLP (WMMA f32): LDS = W1 (128KB) + 8 hid tiles (~130KB)
    size_t shmem = (size_t)TWO_DIM * HIDDEN * sizeof(float)
                 + (size_t)8 * 16 * HSTR * sizeof(float);
    gnn_mlp_kernel<<<(TILES + 7) / 8, 256, shmem, stream>>>(x, W1, b1, W2, b2, out);
}